// ISTA_53154515255813
// MI455X (gfx1250) — compile-verified
//
#include <hip/hip_runtime.h>

// ---- CDNA5 (gfx1250) wave32 WMMA types --------------------------------------
typedef float v2f __attribute__((ext_vector_type(2)));
typedef float v8f __attribute__((ext_vector_type(8)));

// D(16x16,f32) = A(16x4,f32) x B(4x16,f32) + C(16x16,f32)
// 8-arg VOP3P form: (neg_a, A, neg_b, B, c_mod, C, reuse_a, reuse_b)
__device__ __forceinline__ v8f wmma4(v2f a, v2f b, v8f c) {
  return __builtin_amdgcn_wmma_f32_16x16x4_f32(false, a, false, b, (short)0, c,
                                               false, false);
}

// Problem constants (from reference setup_inputs)
#define M_DIM 70     // y cols / W cols
#define N_DIM 100    // S is n x n, d/Wy cols
#define B_ROWS 4096  // batch rows
#define NPAD 112     // 7 tiles of 16 columns
#define KT_S 25      // K-steps of 4 over K=100
#define KT_W 18      // K-steps of 4 over K=70 (padded to 72)
#define WAVES 2      // waves per block
#define THREADS (WAVES * 32)

// LDS:
//  smem: B-operand pair layout: smem[(k2*NPAD + n)*2 + p] = Mat[n][2*k2+p]
//        (first used for W (36 k2-rows), then overwritten with S (50 k2-rows))
//  dbuf: per-wave 16x100 d tile, row-major stride 100
__global__ __launch_bounds__(THREADS) void ISTA_53154515255813_kernel(
    const float* __restrict__ y, const float* __restrict__ S,
    const float* __restrict__ W, const float* __restrict__ thrp,
    const int* __restrict__ iterp, float* __restrict__ out) {
  __shared__ float smem[50 * NPAD * 2];          // 44800 B
  __shared__ float dbuf[WAVES * 16 * N_DIM];     // 12800 B

  const int tid = threadIdx.x;
  const int lane = tid & 31;
  const int wave = tid >> 5;
  const int ln = lane & 15;        // column-in-tile / row-in-tile index
  const int hi = lane >> 4;        // 0: K+0..1, 1: K+2..3 (A) ; k2 parity (B)
  const int row0 = (blockIdx.x * WAVES + wave) * 16;  // this wave's 16 rows

  // ---- Stage W into smem as B-fragment pairs: smem[(k2*NPAD+n)*2+p]=W[n][2k2+p]
  for (int i = tid; i < 36 * NPAD; i += THREADS) {
    const int k2 = i / NPAD, n = i % NPAD;
    const int ka = 2 * k2, kb = ka + 1;
    smem[2 * i + 0] = (n < N_DIM && ka < M_DIM) ? W[n * M_DIM + ka] : 0.0f;
    smem[2 * i + 1] = (n < N_DIM && kb < M_DIM) ? W[n * M_DIM + kb] : 0.0f;
  }
  // Zero this wave's d tile (d0 = 0)
  float* db = &dbuf[wave * 16 * N_DIM];
  for (int i = lane; i < 16 * N_DIM; i += 32) db[i] = 0.0f;
  __syncthreads();

  // ---- Wy = y @ W^T for this wave's 16 rows, kept in registers (C/D layout)
  // A fragments from global y (zero-pad K=70..71)
  v2f ay[KT_W];
  {
    const float* yb = y + (size_t)row0 * M_DIM + (size_t)ln * M_DIM;
#pragma unroll
    for (int k = 0; k < KT_W; ++k) {
      const int colA = 4 * k + 2 * hi;
      v2f a;
      a.x = (colA + 0 < M_DIM) ? yb[colA + 0] : 0.0f;
      a.y = (colA + 1 < M_DIM) ? yb[colA + 1] : 0.0f;
      ay[k] = a;
    }
  }
  v8f wy[7];
#pragma unroll
  for (int j = 0; j < 7; ++j) {
    v8f c = {};
#pragma unroll
    for (int k = 0; k < KT_W; ++k) {
      const v2f b =
          *(const v2f*)&smem[((2 * k + hi) * NPAD + 16 * j + ln) * 2];
      c = wmma4(ay[k], b, c);
    }
    wy[j] = c;
  }
  __syncthreads();

  // ---- Overwrite smem with S^T pairs: smem[(k2*NPAD+n)*2+p] = S[n][2k2+p]
  for (int i = tid; i < 50 * NPAD; i += THREADS) {
    const int k2 = i / NPAD, n = i % NPAD;
    const int ka = 2 * k2;
    smem[2 * i + 0] = (n < N_DIM) ? S[n * N_DIM + ka] : 0.0f;
    smem[2 * i + 1] = (n < N_DIM) ? S[n * N_DIM + ka + 1] : 0.0f;
  }
  __syncthreads();

  const float thr = thrp[0];
  const int nIter = iterp[0];

  // A-fragment base: lane<16 reads (row=ln, cols 4k..4k+1),
  //                  lane>=16 reads (row=ln, cols 4k+2..4k+3) -> ds_load_b64
  const float* arow = db + ln * N_DIM + 2 * hi;

  // ---- Main ISTA loop: d = soft_thr(d @ S^T + Wy), fixed nIter iterations.
  for (int it = 0; it < nIter; ++it) {
    // Load this wave's full 16x100 d tile as A fragments (50 VGPRs)
    v2f a[KT_S];
#pragma unroll
    for (int k = 0; k < KT_S; ++k) a[k] = *(const v2f*)(arow + 4 * k);

#pragma unroll
    for (int j = 0; j < 7; ++j) {
      v8f c = wy[j];  // accumulator starts at Wy tile
#pragma unroll
      for (int k = 0; k < KT_S; ++k) {
        const v2f b =
            *(const v2f*)&smem[((2 * k + hi) * NPAD + 16 * j + ln) * 2];
        c = wmma4(a[k], b, c);
      }
      // soft threshold + scatter back to d tile (C/D layout -> row-major LDS)
      const int col = 16 * j + ln;
#pragma unroll
      for (int r = 0; r < 8; ++r) {
        const float x = c[r];
        const float dn = fmaxf(x - thr, 0.0f) - fmaxf(-x - thr, 0.0f);
        const int rowc = hi ? (r + 8) : r;
        if (col < N_DIM) db[rowc * N_DIM + col] = dn;
      }
    }
  }

  // ---- Write result rows (d tile is contiguous [16][100] = out rows slice)
  {
    float* ob = out + (size_t)row0 * N_DIM;
    for (int i = lane; i < 16 * N_DIM; i += 32) ob[i] = db[i];
  }
}

extern "C" void kernel_launch(void* const* d_in, const int* in_sizes, int n_in,
                              void* d_out, int out_size, void* d_ws,
                              size_t ws_size, hipStream_t stream) {
  const float* y = (const float*)d_in[0];    // [4096,70]
  const float* S = (const float*)d_in[1];    // [100,100]
  const float* W = (const float*)d_in[2];    // [100,70]
  const float* thr = (const float*)d_in[3];  // [1]
  const int* nIt = (const int*)d_in[4];      // [1]
  float* out = (float*)d_out;                // [4096,100]

  const int blocks = B_ROWS / (16 * WAVES);  // 128
  hipLaunchKernelGGL(ISTA_53154515255813_kernel, dim3(blocks), dim3(THREADS),
                     0, stream, y, S, W, thr, nIt, out);
}